// GCN_31525059953266
// MI455X (gfx1250) — compile-verified
//
#include <hip/hip_runtime.h>
#include <math.h>

typedef float v2f __attribute__((ext_vector_type(2)));
typedef float v8f __attribute__((ext_vector_type(8)));

#define N_NODES 100000
#define N_EDGES 1600000
#define D_INF   128
#define D_H     64

// ---------------------------------------------------------------------------
// Degree / normalization kernels
// ---------------------------------------------------------------------------
__global__ __launch_bounds__(256) void deg_init_k(float* __restrict__ deg) {
  int i = blockIdx.x * blockDim.x + threadIdx.x;
  if (i < N_NODES) deg[i] = 1.0f;               // self-loop weight
}

__global__ __launch_bounds__(256) void deg_accum_k(const int* __restrict__ dst,
                                                   const float* __restrict__ w,
                                                   float* __restrict__ deg) {
  int e = blockIdx.x * blockDim.x + threadIdx.x;
  if (e < N_EDGES) atomicAdd(&deg[dst[e]], w[e]);
}

__global__ __launch_bounds__(256) void deg_to_dinv_k(float* __restrict__ deg) {
  int i = blockIdx.x * blockDim.x + threadIdx.x;
  if (i < N_NODES) {
    float v = deg[i];
    deg[i] = (v > 0.0f) ? rsqrtf(v) : 0.0f;
  }
}

__global__ __launch_bounds__(256) void norm_k(const int* __restrict__ src,
                                              const int* __restrict__ dst,
                                              const float* __restrict__ w,
                                              const float* __restrict__ dinv,
                                              float* __restrict__ norm) {
  int e = blockIdx.x * blockDim.x + threadIdx.x;
  if (e < N_EDGES) norm[e] = dinv[src[e]] * w[e] * dinv[dst[e]];
}

// ---------------------------------------------------------------------------
// Dense GEMM: out[N, 64] = A[N, K] @ W[K, 64] using V_WMMA_F32_16X16X4_F32.
// One wave per 16-row tile; each wave computes all four 16x16 column tiles.
//
// W is staged into LDS in a k-pair-interleaved layout:
//   w_lds[(k>>1)*2*D_H + col*2 + (k&1)] = W[k][col]
// so every B fragment { W[kb][col], W[kb+1][col] } is one aligned 8-byte
// ds_load_b64 straight into an even VGPR pair (no register shuffles), and the
// four column-tile fragments are 128 B apart (fusable into ds_load_2addr_b64).
//
// A fragment layout (32-bit A 16x4): lanes 0-15 hold K=k,k+1; lanes 16-31 hold
// K=k+2,k+3 for row = tile*16 + (lane&15).
// ---------------------------------------------------------------------------
template <int K>
__global__ __launch_bounds__(256) void gemm_wmma_k(const float* __restrict__ A,
                                                   const float* __restrict__ W,
                                                   float* __restrict__ out) {
  __shared__ float w_lds[K * D_H];
  // Transposing stage: contiguous global reads, scattered LDS writes.
  for (int i = threadIdx.x; i < K * D_H; i += blockDim.x) {
    int k = i >> 6;            // i / D_H
    int col = i & 63;          // i % D_H
    w_lds[(k >> 1) * (2 * D_H) + col * 2 + (k & 1)] = W[i];
  }
  __syncthreads();

  const int wave      = (blockIdx.x * blockDim.x + threadIdx.x) >> 5;
  const int lane      = threadIdx.x & 31;
  const int row_tile  = wave;                    // 16 rows per wave
  if (row_tile * 16 >= N_NODES) return;

  const int lane_lo   = lane & 15;
  const int khalf     = (lane >> 4) * 2;         // 0 for lanes 0-15, 2 for 16-31
  const int row       = row_tile * 16 + lane_lo;

  // Per-lane base into the interleaved LDS image for this lane's k-half.
  const float* wl = w_lds + (khalf >> 1) * (2 * D_H) + lane_lo * 2;

  v8f acc0 = {}, acc1 = {}, acc2 = {}, acc3 = {};

  for (int k = 0; k < K; k += 4) {
    v2f a = *(const v2f*)(A + (long)row * K + k + khalf);

    const float* wk = wl + (k >> 1) * (2 * D_H);
    v2f b0 = *(const v2f*)(wk +  0);    // col tile 0: cols  0..15
    v2f b1 = *(const v2f*)(wk + 32);    // col tile 1: cols 16..31
    v2f b2 = *(const v2f*)(wk + 64);    // col tile 2: cols 32..47
    v2f b3 = *(const v2f*)(wk + 96);    // col tile 3: cols 48..63

    acc0 = __builtin_amdgcn_wmma_f32_16x16x4_f32(false, a, false, b0, (short)0, acc0, false, false);
    acc1 = __builtin_amdgcn_wmma_f32_16x16x4_f32(false, a, false, b1, (short)0, acc1, false, false);
    acc2 = __builtin_amdgcn_wmma_f32_16x16x4_f32(false, a, false, b2, (short)0, acc2, false, false);
    acc3 = __builtin_amdgcn_wmma_f32_16x16x4_f32(false, a, false, b3, (short)0, acc3, false, false);
  }

  // C/D layout: VGPR r, lanes 0-15 -> M=r; lanes 16-31 -> M=8+r; N=lane&15.
  const int m_base = row_tile * 16 + ((lane >> 4) ? 8 : 0);
#pragma unroll
  for (int r = 0; r < 8; ++r) {
    float* o = out + (long)(m_base + r) * D_H + lane_lo;
    o[ 0] = acc0[r];
    o[16] = acc1[r];
    o[32] = acc2[r];
    o[48] = acc3[r];
  }
}

// ---------------------------------------------------------------------------
// agg[n][d] = hw[n][d] * dinv[n]^2 (self-loop) + bias[d]
// ---------------------------------------------------------------------------
__global__ __launch_bounds__(256) void agg_init_k(const float* __restrict__ hw,
                                                  const float* __restrict__ dinv,
                                                  const float* __restrict__ bias,
                                                  float* __restrict__ agg) {
  int i = blockIdx.x * blockDim.x + threadIdx.x;
  if (i >= N_NODES * D_H) return;
  int n = i >> 6, d = i & 63;
  float di = dinv[n];
  agg[i] = hw[i] * di * di + bias[d];
}

// ---------------------------------------------------------------------------
// Edge scatter: one wave32 per edge; lane l owns features 2l, 2l+1.
// Coalesced float2 gather + f32 global atomics.
// ---------------------------------------------------------------------------
__global__ __launch_bounds__(256) void scatter_k(const float* __restrict__ hw,
                                                 const int* __restrict__ src,
                                                 const int* __restrict__ dst,
                                                 const float* __restrict__ norm,
                                                 float* __restrict__ agg) {
  int e = (blockIdx.x * blockDim.x + threadIdx.x) >> 5;
  if (e >= N_EDGES) return;
  int lane = threadIdx.x & 31;
  int s = src[e], d = dst[e];
  float nv = norm[e];                            // wave-uniform load
  float2 v = *(const float2*)(hw + (long)s * D_H + lane * 2);
  float* o = agg + (long)d * D_H + lane * 2;
  atomicAdd(o,     v.x * nv);
  atomicAdd(o + 1, v.y * nv);
}

// ---------------------------------------------------------------------------
// ELU + deterministic hash dropout (p=0.5, scale 2x), in place.
// ---------------------------------------------------------------------------
__global__ __launch_bounds__(256) void epilogue_k(float* __restrict__ h, int layer) {
  int i = blockIdx.x * blockDim.x + threadIdx.x;
  if (i >= N_NODES * D_H) return;
  float v = h[i];
  v = (v > 0.0f) ? v : (expf(v) - 1.0f);         // ELU(alpha=1)
  unsigned hsh = (unsigned)i * 2654435761u + (unsigned)(layer + 1) * 0x9E3779B9u;
  hsh ^= hsh >> 16; hsh *= 0x85EBCA6Bu; hsh ^= hsh >> 13;
  h[i] = (hsh & 1u) ? v * 2.0f : 0.0f;
}

// ---------------------------------------------------------------------------
// Final linear: out[n] = dot(h[n,:], lin_W) + lin_b
// ---------------------------------------------------------------------------
__global__ __launch_bounds__(256) void final_linear_k(const float* __restrict__ h,
                                                      const float* __restrict__ lw,
                                                      const float* __restrict__ lb,
                                                      float* __restrict__ out) {
  int n = blockIdx.x * blockDim.x + threadIdx.x;
  if (n >= N_NODES) return;
  float acc = 0.0f;
#pragma unroll
  for (int d = 0; d < D_H; d += 4) {
    float4 hv = *(const float4*)(h + (long)n * D_H + d);
    float4 wv = *(const float4*)(lw + d);
    acc += hv.x * wv.x + hv.y * wv.y + hv.z * wv.z + hv.w * wv.w;
  }
  out[n] = acc + lb[0];
}

// ---------------------------------------------------------------------------
// Host launcher
// ---------------------------------------------------------------------------
extern "C" void kernel_launch(void* const* d_in, const int* in_sizes, int n_in,
                              void* d_out, int out_size, void* d_ws, size_t ws_size,
                              hipStream_t stream) {
  const float* x        = (const float*)d_in[0];   // [N, 128]
  const int*   eidx     = (const int*)  d_in[1];   // [2, E]
  const float* eattr    = (const float*)d_in[2];   // [E]
  const float* W1       = (const float*)d_in[3];   // [128, 64]
  const float* b1       = (const float*)d_in[4];   // [64]
  const float* Wh       = (const float*)d_in[5];   // [2, 64, 64]
  const float* bh       = (const float*)d_in[6];   // [2, 64]
  const float* lin_W    = (const float*)d_in[7];   // [64]
  const float* lin_b    = (const float*)d_in[8];   // [1]
  float*       out      = (float*)d_out;           // [N]

  const int* src = eidx;
  const int* dst = eidx + N_EDGES;

  // Workspace layout (floats): dinv[N] | norm[E] | bufA[N*64] | bufB[N*64]
  float* dinv = (float*)d_ws;
  float* norm = dinv + N_NODES;
  float* bufA = norm + N_EDGES;
  float* bufB = bufA + (size_t)N_NODES * D_H;

  const int nodeB   = (N_NODES + 255) / 256;
  const int edgeB   = (N_EDGES + 255) / 256;
  const int featB   = (N_NODES * D_H + 255) / 256;
  const int gemmB   = ((N_NODES / 16) * 32 + 255) / 256;   // 1 wave / 16 rows
  const int scatB   = ((long)N_EDGES * 32 + 255) / 256;    // 1 wave / edge

  // ---- gcn_norm ----
  deg_init_k   <<<nodeB, 256, 0, stream>>>(dinv);
  deg_accum_k  <<<edgeB, 256, 0, stream>>>(dst, eattr, dinv);
  deg_to_dinv_k<<<nodeB, 256, 0, stream>>>(dinv);
  norm_k       <<<edgeB, 256, 0, stream>>>(src, dst, eattr, dinv, norm);

  // ---- layer 1: x @ W1 -> bufA; aggregate -> bufB; elu+dropout ----
  gemm_wmma_k<D_INF><<<gemmB, 256, 0, stream>>>(x, W1, bufA);
  agg_init_k  <<<featB, 256, 0, stream>>>(bufA, dinv, b1, bufB);
  scatter_k   <<<scatB, 256, 0, stream>>>(bufA, src, dst, norm, bufB);
  epilogue_k  <<<featB, 256, 0, stream>>>(bufB, 0);

  // ---- hidden layers 2,3 ----
  for (int l = 0; l < 2; ++l) {
    gemm_wmma_k<D_H><<<gemmB, 256, 0, stream>>>(bufB, Wh + (size_t)l * D_H * D_H, bufA);
    agg_init_k  <<<featB, 256, 0, stream>>>(bufA, dinv, bh + (size_t)l * D_H, bufB);
    scatter_k   <<<scatB, 256, 0, stream>>>(bufA, src, dst, norm, bufB);
    epilogue_k  <<<featB, 256, 0, stream>>>(bufB, l + 1);
  }

  // ---- final linear ----
  final_linear_k<<<nodeB, 256, 0, stream>>>(bufB, lin_W, lin_b, out);
}